// VectorQuantizer_55301998903736
// MI455X (gfx1250) — compile-verified
//
#include <hip/hip_runtime.h>

typedef __attribute__((ext_vector_type(2))) float v2f;
typedef __attribute__((ext_vector_type(8))) float v8f;

#define N_ELEM   16777216ull     // 64^4
#define DIM      64
#define KCODES   512
#define WAVES_PB 8
#define TPB      256
#define RSETS    2               // 16-row tiles per wave (32 rows/wave)
#define NBLOCKS  1024            // 262144 rows / (32 rows * 8 waves)

// ---- column norms ||e_k||^2, k = 0..511 (one tiny block, once per launch)
__global__ void vq_colnorm_kernel(const float* __restrict__ emb,
                                  float* __restrict__ norms) {
    int c = threadIdx.x;         // 512 threads
    float s = 0.f;
    #pragma unroll
    for (int d = 0; d < DIM; ++d) {
        float e = emb[d * KCODES + c];
        s += e * e;
    }
    norms[c] = s;
}

// ---- main kernel: f32 WMMA distance tiles + fused argmin + gather + loss
__global__ __launch_bounds__(TPB) void vq_main_kernel(
        const float* __restrict__ x, const float* __restrict__ emb,
        const float* __restrict__ norms, float* __restrict__ out,
        float* __restrict__ partial) {
    __shared__ float s_red[WAVES_PB];

    const int t    = threadIdx.x;
    const int wave = t >> 5;
    const int lane = t & 31;
    const int row0 = (blockIdx.x * WAVES_PB + wave) * (16 * RSETS);

    const int m  = lane & 15;        // A: row in tile; B/C: column in tile
    const int kk = (lane >> 4) * 2;  // K sub-offset (lanes 0-15 -> K0/1, 16-31 -> K2/3)

    // ---- A fragments for RSETS 16-row tiles; x read exactly once, nontemporal
    v2f a[RSETS][16];
    #pragma unroll
    for (int s = 0; s < RSETS; ++s) {
        const float* xrow = x + (size_t)(row0 + s * 16 + m) * DIM;
        #pragma unroll
        for (int kf = 0; kf < 16; ++kf)
            a[s][kf] = __builtin_nontemporal_load((const v2f*)(xrow + kf * 4 + kk));
    }

    float bestd[RSETS][8];
    int   bestc[RSETS][8];
    #pragma unroll
    for (int s = 0; s < RSETS; ++s)
        #pragma unroll
        for (int i = 0; i < 8; ++i) { bestd[s][i] = 3.4e38f; bestc[s][i] = 0; }

    for (int tile = 0; tile < 32; ++tile) {
        const int cbase = tile * 16;
        v8f acc0 = {};               // row set 0
        v8f acc1 = {};               // row set 1 (independent WMMA chain)
        #pragma unroll
        for (int kf = 0; kf < 16; ++kf) {
            const int kb = kf * 4;
            v2f b;                   // one B fragment feeds both row sets
            b.x = emb[(kb + kk)     * KCODES + cbase + m];
            b.y = emb[(kb + kk + 1) * KCODES + cbase + m];
            acc0 = __builtin_amdgcn_wmma_f32_16x16x4_f32(
                       false, a[0][kf], false, b, (short)0, acc0, false, false);
            acc1 = __builtin_amdgcn_wmma_f32_16x16x4_f32(
                       false, a[1][kf], false, b, (short)0, acc1, false, false);
        }
        const float cn = norms[cbase + m];
        #pragma unroll
        for (int i = 0; i < 8; ++i) {
            float d0 = cn - 2.0f * acc0[i];   // ||z||^2 dropped: constant per row
            float d1 = cn - 2.0f * acc1[i];
            if (d0 < bestd[0][i]) { bestd[0][i] = d0; bestc[0][i] = cbase + m; }
            if (d1 < bestd[1][i]) { bestd[1][i] = d1; bestc[1][i] = cbase + m; }
        }
    }

    // Cross-lane argmin within each 16-lane half (VGPR i holds rows i / i+8).
    // Tie-break on lower column index to match jnp.argmin first-occurrence.
    #pragma unroll
    for (int off = 8; off >= 1; off >>= 1) {
        #pragma unroll
        for (int s = 0; s < RSETS; ++s) {
            #pragma unroll
            for (int i = 0; i < 8; ++i) {
                float od = __shfl_xor(bestd[s][i], off, 32);
                int   oc = __shfl_xor(bestc[s][i], off, 32);
                if (od < bestd[s][i] || (od == bestd[s][i] && oc < bestc[s][i])) {
                    bestd[s][i] = od; bestc[s][i] = oc;
                }
            }
        }
    }
    // Now every lane of a half holds identical bests: lower half rows 0-7,
    // upper half rows 8-15 (per row set).

    float sq = 0.f;
    #pragma unroll
    for (int s = 0; s < RSETS; ++s) {
        // ---- per-lane winning index for *its own* row m (for loss from regs)
        int c01 = (m & 1) ? bestc[s][1] : bestc[s][0];
        int c23 = (m & 1) ? bestc[s][3] : bestc[s][2];
        int c45 = (m & 1) ? bestc[s][5] : bestc[s][4];
        int c67 = (m & 1) ? bestc[s][7] : bestc[s][6];
        int c03 = (m & 2) ? c23 : c01;
        int c47 = (m & 2) ? c67 : c45;
        int cm  = (m & 4) ? c47 : c03;             // = bestc[s][m&7] on this lane
        int idx_m = __shfl(cm, (m < 8) ? m : (m + 8), 32);  // row m's index

        // squared error from A registers (no x re-read from HBM)
        #pragma unroll
        for (int kf = 0; kf < 16; ++kf) {
            const int d = kf * 4 + kk;
            float q0 = emb[(d    ) * KCODES + idx_m];
            float q1 = emb[(d + 1) * KCODES + idx_m];
            float e0 = q0 - a[s][kf].x;
            float e1 = q1 - a[s][kf].y;
            sq += e0 * e0 + e1 * e1;
        }

        // ---- coalesced quantized writes (row broadcast, lanes span columns)
        #pragma unroll
        for (int j = 0; j < 16; ++j) {
            int idx = __shfl(bestc[s][j & 7], (j < 8) ? 0 : 16, 32);
            const size_t r = (size_t)(row0 + s * 16 + j) * DIM;
            float q0 = emb[(size_t)lane        * KCODES + idx];
            float q1 = emb[(size_t)(lane + 32) * KCODES + idx];
            __builtin_nontemporal_store(q0, out + r + lane);
            __builtin_nontemporal_store(q1, out + r + lane + 32);
        }
    }

    #pragma unroll
    for (int off = 16; off >= 1; off >>= 1)
        sq += __shfl_xor(sq, off, 32);
    if (lane == 0) s_red[wave] = sq;
    __syncthreads();
    if (t == 0) {
        float s = 0.f;
        #pragma unroll
        for (int w = 0; w < WAVES_PB; ++w) s += s_red[w];
        partial[blockIdx.x] = s;
    }
}

// ---- deterministic final loss reduction (no float atomics)
__global__ void vq_loss_kernel(const float* __restrict__ partial,
                               float* __restrict__ loss) {
    __shared__ float s_red[TPB / 32];
    float acc = 0.f;
    for (int i = threadIdx.x; i < NBLOCKS; i += TPB) acc += partial[i];
    #pragma unroll
    for (int off = 16; off >= 1; off >>= 1) acc += __shfl_xor(acc, off, 32);
    if ((threadIdx.x & 31) == 0) s_red[threadIdx.x >> 5] = acc;
    __syncthreads();
    if (threadIdx.x == 0) {
        float tot = 0.f;
        #pragma unroll
        for (int w = 0; w < TPB / 32; ++w) tot += s_red[w];
        *loss = tot * (1.25f / (float)N_ELEM);   // (BETA + 1) * mean
    }
}

extern "C" void kernel_launch(void* const* d_in, const int* in_sizes, int n_in,
                              void* d_out, int out_size, void* d_ws, size_t ws_size,
                              hipStream_t stream) {
    const float* x   = (const float*)d_in[0];   // [64,64,64,64] f32
    const float* emb = (const float*)d_in[1];   // [64,512] f32
    float* out       = (float*)d_out;           // quantized (16777216) + loss (1)
    float* partial   = (float*)d_ws;            // NBLOCKS floats
    float* norms     = partial + NBLOCKS;       // KCODES floats

    vq_colnorm_kernel<<<1, KCODES, 0, stream>>>(emb, norms);
    vq_main_kernel<<<NBLOCKS, TPB, 0, stream>>>(x, emb, norms, out, partial);
    vq_loss_kernel<<<1, TPB, 0, stream>>>(partial, out + N_ELEM);
}